// GSAN_35433480192470
// MI455X (gfx1250) — compile-verified
//
#include <hip/hip_runtime.h>
#include <math.h>

// ---------------- problem constants (from reference) ----------------
#define NNODES   50000
#define NDIM     128
#define NEDGES   800000
#define NLEV     3
#define NSTEPS   3
#define NHEADS   4
#define NFEATS   16
#define HF       (NHEADS * NFEATS)        // 64
#define NCOLS    (NLEV * HF)              // 192
#define LEAKY    0.2f

typedef float v2f __attribute__((ext_vector_type(2)));
typedef float v8f __attribute__((ext_vector_type(8)));

// ---------------- helpers ----------------
__device__ __forceinline__ unsigned flip_f(float f) {
    unsigned u = __float_as_uint(f);
    // monotonic float -> uint mapping
    return u ^ (unsigned)(((int)u >> 31) | 0x80000000);
}
__device__ __forceinline__ float unflip_f(unsigned u) {
    unsigned m = (u >> 31) ? 0x80000000u : 0xFFFFFFFFu;
    return __uint_as_float(u ^ m);
}

// ---------------- zero scratch ----------------
__global__ __launch_bounds__(256) void k_zero(float* __restrict__ p, int n) {
    int i = blockIdx.x * 256 + threadIdx.x;
    if (i < n) p[i] = 0.0f;
}

// ---------------- per-node attention logit projections ----------------
// s[n][h] = x[n] . src_kernel[:,h] ; t[n][h] = x[n] . trg_kernel[:,h]
__global__ __launch_bounds__(256) void k_node_logits(
    const float* __restrict__ x, const float* __restrict__ srcK,
    const float* __restrict__ trgK, float* __restrict__ s, float* __restrict__ t) {
    int tid = blockIdx.x * 256 + threadIdx.x;
    if (tid >= NNODES * 8) return;
    int n = tid >> 3, r = tid & 7, h = r & 3;
    const float* K  = (r < 4) ? srcK : trgK;
    const float* xr = x + n * NDIM;
    float acc = 0.0f;
#pragma unroll 4
    for (int d = 0; d < NDIM; ++d) acc += xr[d] * K[d * NHEADS + h];
    if (r < 4) s[n * 4 + h] = acc;
    else       t[n * 4 + h] = acc;
}

// ---------------- u = x @ W + bias via V_WMMA_F32_16X16X4_F32 ----------------
// One wave32 per 16-row M tile; all 12 N-tiles (192 cols) accumulated in
// registers so x is streamed exactly once. K walked in chunks of 4.
// kernel layout: kernel[l][d][h][f] flat = l*8192 + d*64 + h*16 + f
__global__ __launch_bounds__(32) void k_u_wmma(
    const float* __restrict__ x, const float* __restrict__ kern,
    const float* __restrict__ bias, float* __restrict__ u) {
    const int lane  = threadIdx.x;          // 0..31, EXEC all ones
    const int m0    = blockIdx.x * 16;      // 3125 * 16 == 50000 exactly
    const int mrow  = m0 + (lane & 15);
    const int kpair = (lane >> 4) * 2;      // A/B: lanes 0-15 hold K+0/K+1, 16-31 hold K+2/K+3

    v8f acc[12];
    const v8f vz = {0.f, 0.f, 0.f, 0.f, 0.f, 0.f, 0.f, 0.f};
#pragma unroll
    for (int nt = 0; nt < 12; ++nt) acc[nt] = vz;

    for (int k0 = 0; k0 < NDIM; k0 += 4) {
        const int r = k0 + kpair;
        v2f a;
        a.x = x[mrow * NDIM + r];
        a.y = x[mrow * NDIM + r + 1];
#pragma unroll
        for (int nt = 0; nt < 12; ++nt) {
            const int c = nt * 16 + (lane & 15);          // global output column
            const int l = c >> 6, h = (c >> 4) & 3, f = c & 15;
            const int base = l * (NDIM * HF) + h * NFEATS + f;
            v2f b;
            b.x = kern[base + r * HF];
            b.y = kern[base + (r + 1) * HF];
            acc[nt] = __builtin_amdgcn_wmma_f32_16x16x4_f32(
                false, a, false, b, (short)0, acc[nt], false, false);
        }
    }
    // C/D layout: VGPR j -> lanes 0-15 row m0+j, lanes 16-31 row m0+8+j; N = lane&15
#pragma unroll
    for (int nt = 0; nt < 12; ++nt) {
        const int c  = nt * 16 + (lane & 15);
        const float bv = bias[c];
#pragma unroll
        for (int j = 0; j < 8; ++j) {
            const int row = m0 + j + 8 * (lane >> 4);
            u[row * NCOLS + c] = acc[nt][j] + bv;
        }
    }
}

// ---------------- segment softmax: pass 1 (max) ----------------
__global__ __launch_bounds__(256) void k_edge_max(
    const int* __restrict__ eidx, const float* __restrict__ s,
    const float* __restrict__ t, unsigned* __restrict__ segmax) {
    int tid = blockIdx.x * 256 + threadIdx.x;
    if (tid >= NEDGES * 4) return;
    int e = tid >> 2, h = tid & 3;
    int rcv = eidx[e * 2 + 0], snd = eidx[e * 2 + 1];
    float lg = s[snd * 4 + h] + t[rcv * 4 + h];
    lg = (lg >= 0.0f) ? lg : LEAKY * lg;
    atomicMax(&segmax[rcv * 4 + h], flip_f(lg));
}

// ---------------- segment softmax: pass 2 (exp + sum) ----------------
__global__ __launch_bounds__(256) void k_edge_expsum(
    const int* __restrict__ eidx, const float* __restrict__ s,
    const float* __restrict__ t, const unsigned* __restrict__ segmax,
    float* __restrict__ attn, float* __restrict__ ssum) {
    int tid = blockIdx.x * 256 + threadIdx.x;
    if (tid >= NEDGES * 4) return;
    int e = tid >> 2, h = tid & 3;
    int rcv = eidx[e * 2 + 0], snd = eidx[e * 2 + 1];
    float lg = s[snd * 4 + h] + t[rcv * 4 + h];
    lg = (lg >= 0.0f) ? lg : LEAKY * lg;
    float ev = __expf(lg - unflip_f(segmax[rcv * 4 + h]));
    attn[e * 4 + h] = ev;
    atomicAdd(&ssum[rcv * 4 + h], ev);
}

// ---------------- segment softmax: pass 3 (normalize) ----------------
__global__ __launch_bounds__(256) void k_attn_norm(
    const int* __restrict__ eidx, const float* __restrict__ ssum,
    float* __restrict__ attn) {
    int tid = blockIdx.x * 256 + threadIdx.x;
    if (tid >= NEDGES * 4) return;
    int e = tid >> 2, h = tid & 3;
    int rcv = eidx[e * 2 + 0];
    attn[tid] = attn[tid] / (ssum[rcv * 4 + h] + 1e-16f);
}

// ---------------- F0 = cumprod(u, axis=level)  (BASEPOINT) ----------------
__global__ __launch_bounds__(256) void k_initF(
    const float* __restrict__ u, float* __restrict__ F) {
    int tid = blockIdx.x * 256 + threadIdx.x;
    if (tid >= NNODES * HF) return;
    int n = tid >> 6, c = tid & 63;
    const float* ur = u + n * NCOLS + c;
    float u0 = ur[0], u1 = ur[HF], u2 = ur[2 * HF];
    float* Fr = F + n * NCOLS + c;
    float f0 = u0, f1 = u0 * u1, f2 = f1 * u2;
    Fr[0] = f0; Fr[HF] = f1; Fr[2 * HF] = f2;
}

// ---------------- one message-passing step (scatter-add) ----------------
// thread = (edge, hf-column). Covers all 3 levels so the gathered g values
// double as the "lower" shifted terms (DIFF path: inc = u[snd]-u[rcv]).
__global__ __launch_bounds__(256) void k_step(
    const int* __restrict__ eidx, const float* __restrict__ attn,
    const float* __restrict__ u, const float* __restrict__ Fin,
    float* __restrict__ Fout) {
    int tid = blockIdx.x * 256 + threadIdx.x;
    if (tid >= NEDGES * HF) return;
    int e = tid >> 6, c = tid & 63, h = c >> 4;
    int rcv = eidx[e * 2 + 0], snd = eidx[e * 2 + 1];
    float a = attn[e * 4 + h];
    const float* gs = Fin + snd * NCOLS + c;
    const float* us = u + snd * NCOLS + c;
    const float* ur = u + rcv * NCOLS + c;
    float g0 = gs[0], g1 = gs[HF], g2 = gs[2 * HF];
    float i0 = us[0] - ur[0];
    float i1 = us[HF] - ur[HF];
    float i2 = us[2 * HF] - ur[2 * HF];
    float m0 = (g0 + i0) * a;         // lower = 1 at level 0
    float m1 = (g1 + g0 * i1) * a;
    float m2 = (g2 + g1 * i2) * a;
    float* out = Fout + rcv * NCOLS + c;
    atomicAdd(out,          m0);
    atomicAdd(out + HF,     m1);
    atomicAdd(out + 2 * HF, m2);
}

// ---------------- feat = embed_coeffs * F ----------------
__global__ __launch_bounds__(256) void k_final(
    const float* __restrict__ F, const float* __restrict__ emb,
    float* __restrict__ out) {
    int tid = blockIdx.x * 256 + threadIdx.x;
    if (tid >= NNODES * NCOLS) return;
    out[tid] = emb[tid % NCOLS] * F[tid];
}

// ---------------- host-side orchestration ----------------
static inline int cdiv(long a, long b) { return (int)((a + b - 1) / b); }

extern "C" void kernel_launch(void* const* d_in, const int* in_sizes, int n_in,
                              void* d_out, int out_size, void* d_ws, size_t ws_size,
                              hipStream_t stream) {
    (void)in_sizes; (void)n_in; (void)out_size; (void)ws_size;
    const float* x    = (const float*)d_in[0];
    const int*   eidx = (const int*)  d_in[1];
    const float* kern = (const float*)d_in[2];
    const float* srcK = (const float*)d_in[3];
    const float* trgK = (const float*)d_in[4];
    const float* bias = (const float*)d_in[5];
    const float* emb  = (const float*)d_in[6];
    float* out = (float*)d_out;

    // workspace carve-up (floats)
    float* w    = (float*)d_ws;
    float* u    = w;                                   // 50000*192
    float* FA   = u  + (long)NNODES * NCOLS;           // 50000*192
    float* FB   = FA + (long)NNODES * NCOLS;           // 50000*192
    float* s    = FB + (long)NNODES * NCOLS;           // 50000*4
    float* t    = s  + NNODES * 4;                     // 50000*4
    float* ssum = t  + NNODES * 4;                     // 50000*4
    float* attn = ssum + NNODES * 4;                   // 800000*4
    unsigned* segmax = (unsigned*)(attn + (long)NEDGES * 4); // 50000*4

    const int NH = NNODES * 4;
    const int EH = NEDGES * 4;
    const int NF = NNODES * HF;
    const int EF = NEDGES * HF;
    const int NC = NNODES * NCOLS;

    // init scratch that is accumulated into
    k_zero<<<cdiv(NH, 256), 256, 0, stream>>>((float*)segmax, NH);
    k_zero<<<cdiv(NH, 256), 256, 0, stream>>>(ssum, NH);

    // projections + u GEMM (WMMA)
    k_node_logits<<<cdiv(NNODES * 8, 256), 256, 0, stream>>>(x, srcK, trgK, s, t);
    k_u_wmma<<<NNODES / 16, 32, 0, stream>>>(x, kern, bias, u);

    // segment softmax over receivers
    k_edge_max   <<<cdiv(EH, 256), 256, 0, stream>>>(eidx, s, t, segmax);
    k_edge_expsum<<<cdiv(EH, 256), 256, 0, stream>>>(eidx, s, t, segmax, attn, ssum);
    k_attn_norm  <<<cdiv(EH, 256), 256, 0, stream>>>(eidx, ssum, attn);

    // F0 = cumprod(u)
    k_initF<<<cdiv(NF, 256), 256, 0, stream>>>(u, FA);

    // three scan steps, ping-pong FA <-> FB
    k_zero<<<cdiv(NC, 256), 256, 0, stream>>>(FB, NC);
    k_step<<<cdiv(EF, 256), 256, 0, stream>>>(eidx, attn, u, FA, FB);
    k_zero<<<cdiv(NC, 256), 256, 0, stream>>>(FA, NC);
    k_step<<<cdiv(EF, 256), 256, 0, stream>>>(eidx, attn, u, FB, FA);
    k_zero<<<cdiv(NC, 256), 256, 0, stream>>>(FB, NC);
    k_step<<<cdiv(EF, 256), 256, 0, stream>>>(eidx, attn, u, FA, FB);

    // feat = embed_coeffs * F
    k_final<<<cdiv(NC, 256), 256, 0, stream>>>(FB, emb, out);
}